// PlasticConv1d_85512798863598
// MI455X (gfx1250) — compile-verified
//
#include <hip/hip_runtime.h>

typedef __attribute__((ext_vector_type(2))) float v2f;
typedef __attribute__((ext_vector_type(8))) float v8f;

#define BB   64
#define CIN  256
#define COUT 256
#define TT   2048
#define ETA  0.1f

// ---------------------------------------------------------------------------
// Kernel 1: transpose x (B, Cin, T) -> xT (B, T, Cin)   [32x32 LDS tiles]
// ---------------------------------------------------------------------------
__global__ void k_transpose_x(const float* __restrict__ x, float* __restrict__ xT) {
  __shared__ float tile[32][33];
  int b  = blockIdx.z;
  int t0 = blockIdx.x * 32;
  int c0 = blockIdx.y * 32;
  int tx = threadIdx.x, ty = threadIdx.y;
  const float* xb = x  + (size_t)b * CIN * TT;
  float*       ob = xT + (size_t)b * TT * CIN;
#pragma unroll
  for (int k = 0; k < 4; k++) {
    int c = c0 + ty + k * 8;
    tile[ty + k * 8][tx] = xb[(size_t)c * TT + (t0 + tx)];
  }
  __syncthreads();
#pragma unroll
  for (int k = 0; k < 4; k++) {
    int t = t0 + ty + k * 8;
    ob[(size_t)t * CIN + (c0 + tx)] = tile[tx][ty + k * 8];
  }
}

// ---------------------------------------------------------------------------
// Kernel 2: ystatic[(b,t), o] = xT[(b,t), :] . W[o, :] + bias[o]
// WMMA f32 16x16x4. Each wave owns one 16(M) x 16(N) tile, K marched by 4.
// A layout (16x4 f32): lane m<16 holds (K=k0,k0+1) in v[0],v[1]; lanes 16-31
// hold (k0+2,k0+3)  -> one b64 load per lane. B (4x16) mirrors this with
// B[k][n] = W[n][k] -> one b64 load per lane from W row n.
// ---------------------------------------------------------------------------
__global__ void __launch_bounds__(256) k_gemm_wmma(
    const float* __restrict__ xT, const float* __restrict__ W,
    const float* __restrict__ bias, float* __restrict__ yst) {
  int l    = threadIdx.x & 31;
  int w    = threadIdx.x >> 5;      // wave 0..7
  int half = l >> 4;                // 0: K pair (0,1), 1: K pair (2,3)
  int lm   = l & 15;
  int m0   = blockIdx.x * 128 + w * 16;   // row in (B*T)
  int n0   = blockIdx.y * 16;             // output channel tile
  const float* Arow = xT + (size_t)(m0 + lm) * CIN;
  const float* Brow = W  + (size_t)(n0 + lm) * CIN;
  v8f acc = {0.f, 0.f, 0.f, 0.f, 0.f, 0.f, 0.f, 0.f};
#pragma unroll 8
  for (int k0 = 0; k0 < CIN; k0 += 4) {
    v2f a  = *(const v2f*)(Arow + k0 + half * 2);
    v2f bv = *(const v2f*)(Brow + k0 + half * 2);
    acc = __builtin_amdgcn_wmma_f32_16x16x4_f32(
        /*neg_a=*/false, a, /*neg_b=*/false, bv,
        /*c_mod=*/(short)0, acc, /*reuse_a=*/false, /*reuse_b=*/false);
  }
  float bs = bias[n0 + lm];
  // D layout: VGPR r, lane l -> M = r + 8*half, N = lm
#pragma unroll
  for (int r = 0; r < 8; r++) {
    yst[(size_t)(m0 + r + half * 8) * COUT + (n0 + lm)] = acc[r] + bs;
  }
}

// ---------------------------------------------------------------------------
// Kernel 3: sequential Oja recurrence. Rows (b,o) are independent given x_t.
// 8 lanes per row; each lane owns h[32] in VGPRs. No LDS, no barriers.
//   y      = ystatic + h . x_t          (3 shfl_xor reduce inside wave32)
//   h      = (1 - eta*m*y^2) * h + eta*m*y * x_t
// ybuf holds ystatic on entry and is overwritten with y_t in place.
// ---------------------------------------------------------------------------
#define LPR         8
#define RTHREADS    512
#define ROWS_PER_WG (RTHREADS / LPR)   // 64
#define WG_PER_B    (COUT / ROWS_PER_WG) // 4

__global__ void __launch_bounds__(RTHREADS, 1) k_recur(
    const float* __restrict__ xT, const float* __restrict__ mods,
    float* __restrict__ ybuf) {
  int tid = threadIdx.x;
  int sub = tid & (LPR - 1);          // 32-wide chunk of Cin
  int rl  = tid >> 3;                 // local row 0..63
  int wg  = blockIdx.x;
  int b   = wg / WG_PER_B;
  int o   = (wg % WG_PER_B) * ROWS_PER_WG + rl;
  const float* xb = xT   + (size_t)b * TT * CIN + sub * 32;
  const float* mb = mods + (size_t)b * TT;
  float*       yb = ybuf + (size_t)b * TT * COUT;

  float h[32];
#pragma unroll
  for (int j = 0; j < 32; j++) h[j] = 0.f;

  for (int t = 0; t < TT; t++) {
    const float4* xp = (const float4*)(xb + (size_t)t * CIN);
    float xv[32];
#pragma unroll
    for (int q = 0; q < 8; q++) {
      float4 v = xp[q];
      xv[4 * q + 0] = v.x; xv[4 * q + 1] = v.y;
      xv[4 * q + 2] = v.z; xv[4 * q + 3] = v.w;
    }
    if (t + 4 < TT) __builtin_prefetch(xb + (size_t)(t + 4) * CIN, 0, 0);

    float a0 = 0.f, a1 = 0.f, a2 = 0.f, a3 = 0.f;
#pragma unroll
    for (int j = 0; j < 32; j += 4) {
      a0 = fmaf(h[j + 0], xv[j + 0], a0);
      a1 = fmaf(h[j + 1], xv[j + 1], a1);
      a2 = fmaf(h[j + 2], xv[j + 2], a2);
      a3 = fmaf(h[j + 3], xv[j + 3], a3);
    }
    float acc = (a0 + a1) + (a2 + a3);
    acc += __shfl_xor(acc, 1, 32);
    acc += __shfl_xor(acc, 2, 32);
    acc += __shfl_xor(acc, 4, 32);

    float y     = acc + yb[(size_t)t * COUT + o];
    float m     = mb[t];
    float coef  = ETA * m * y;
    float decay = 1.0f - coef * y;
#pragma unroll
    for (int j = 0; j < 32; j++) h[j] = fmaf(decay, h[j], coef * xv[j]);

    if (sub == 0) yb[(size_t)t * COUT + o] = y;
  }
}

// ---------------------------------------------------------------------------
// Kernel 4: transpose ybuf (B, T, Cout) -> out (B, Cout, T)
// ---------------------------------------------------------------------------
__global__ void k_transpose_y(const float* __restrict__ ybuf, float* __restrict__ out) {
  __shared__ float tile[32][33];
  int b  = blockIdx.z;
  int t0 = blockIdx.x * 32;
  int c0 = blockIdx.y * 32;
  int tx = threadIdx.x, ty = threadIdx.y;
  const float* yb = ybuf + (size_t)b * TT * COUT;
  float*       ob = out  + (size_t)b * COUT * TT;
#pragma unroll
  for (int k = 0; k < 4; k++) {
    int t = t0 + ty + k * 8;
    tile[ty + k * 8][tx] = yb[(size_t)t * COUT + (c0 + tx)];
  }
  __syncthreads();
#pragma unroll
  for (int k = 0; k < 4; k++) {
    int c = c0 + ty + k * 8;
    ob[(size_t)c * TT + (t0 + tx)] = tile[tx][ty + k * 8];
  }
}

// ---------------------------------------------------------------------------
extern "C" void kernel_launch(void* const* d_in, const int* in_sizes, int n_in,
                              void* d_out, int out_size, void* d_ws, size_t ws_size,
                              hipStream_t stream) {
  const float* x    = (const float*)d_in[0];  // (B, Cin, T)
  const float* mods = (const float*)d_in[1];  // (B, T)
  const float* W    = (const float*)d_in[2];  // (Cout, Cin)
  const float* bias = (const float*)d_in[3];  // (Cout,)
  float* out = (float*)d_out;                 // (B, Cout, T)

  float* xT   = (float*)d_ws;                       // (B, T, Cin)  128 MB
  float* ybuf = xT + (size_t)BB * TT * CIN;         // (B, T, Cout) 128 MB

  dim3 tb(32, 8);
  k_transpose_x<<<dim3(TT / 32, CIN / 32, BB), tb, 0, stream>>>(x, xT);
  k_gemm_wmma<<<dim3((BB * TT) / 128, COUT / 16), 256, 0, stream>>>(xT, W, bias, ybuf);
  k_recur<<<BB * WG_PER_B, RTHREADS, 0, stream>>>(xT, mods, ybuf);
  k_transpose_y<<<dim3(TT / 32, COUT / 32, BB), tb, 0, stream>>>(ybuf, out);
}